// ObjectRecognitionNetwork_73547019976741
// MI455X (gfx1250) — compile-verified
//
#include <hip/hip_runtime.h>
#include <hip/hip_bf16.h>
#include <math.h>

typedef __attribute__((ext_vector_type(2))) float v2f;
typedef __attribute__((ext_vector_type(8))) float v8f;

#define NPTS 4096
#define GPTS 64
#define KNN  10

// ---------------------------------------------------------------------------
// Kernel 1: KNN (10-NN over 4096 pts) + covariance eigen features for the
// first 64 points of each batch. 512 blocks (b*64+n), 64 threads each.
// ---------------------------------------------------------------------------
__global__ __launch_bounds__(64) void knn_geom_kernel(const float* __restrict__ pc,
                                                      float* __restrict__ gf) {
    int task = blockIdx.x;            // 0..511
    int b = task >> 6;
    int n = task & 63;
    int t = threadIdx.x;              // 0..63
    const float* base = pc + (size_t)b * NPTS * 3;
    float qx = base[n * 3 + 0], qy = base[n * 3 + 1], qz = base[n * 3 + 2];

    float dist[KNN];
    int   idx[KNN];
#pragma unroll
    for (int i = 0; i < KNN; ++i) { dist[i] = INFINITY; idx[i] = 0x7fffffff; }

    for (int j = t; j < NPTS; j += 64) {
        float dx = base[j * 3 + 0] - qx;
        float dy = base[j * 3 + 1] - qy;
        float dz = base[j * 3 + 2] - qz;
        float d = dx * dx + dy * dy + dz * dz;
        if (d < dist[KNN - 1] || (d == dist[KNN - 1] && j < idx[KNN - 1])) {
            dist[KNN - 1] = d; idx[KNN - 1] = j;
#pragma unroll
            for (int i = KNN - 1; i > 0; --i) {
                bool sw = (dist[i] < dist[i - 1]) ||
                          (dist[i] == dist[i - 1] && idx[i] < idx[i - 1]);
                if (sw) {
                    float td = dist[i]; dist[i] = dist[i - 1]; dist[i - 1] = td;
                    int   ti = idx[i];  idx[i]  = idx[i - 1];  idx[i - 1]  = ti;
                }
            }
        }
    }

    __shared__ float ld[64 * KNN];
    __shared__ int   li[64 * KNN];
#pragma unroll
    for (int i = 0; i < KNN; ++i) { ld[t * KNN + i] = dist[i]; li[t * KNN + i] = idx[i]; }
    __syncthreads();

    if (t == 0) {
        // 64-way merge of sorted per-thread lists -> global 10 smallest
        int head[64];
        for (int w = 0; w < 64; ++w) head[w] = 0;
        float nx[KNN], ny[KNN], nz[KNN];
        for (int k = 0; k < KNN; ++k) {
            int bw = 0; float bd = INFINITY; int bi = 0x7fffffff;
            for (int w = 0; w < 64; ++w) {
                if (head[w] < KNN) {
                    float d = ld[w * KNN + head[w]];
                    int   ii = li[w * KNN + head[w]];
                    if (d < bd || (d == bd && ii < bi)) { bd = d; bi = ii; bw = w; }
                }
            }
            head[bw]++;
            nx[k] = base[bi * 3 + 0];
            ny[k] = base[bi * 3 + 1];
            nz[k] = base[bi * 3 + 2];
        }
        // mean
        float mx = 0.f, my = 0.f, mz = 0.f;
        for (int k = 0; k < KNN; ++k) { mx += nx[k]; my += ny[k]; mz += nz[k]; }
        mx *= 0.1f; my *= 0.1f; mz *= 0.1f;
        // covariance (symmetric 3x3), divided by K
        float c00 = 0.f, c01 = 0.f, c02 = 0.f, c11 = 0.f, c12 = 0.f, c22 = 0.f;
        for (int k = 0; k < KNN; ++k) {
            float dx = nx[k] - mx, dy = ny[k] - my, dz = nz[k] - mz;
            c00 += dx * dx; c01 += dx * dy; c02 += dx * dz;
            c11 += dy * dy; c12 += dy * dz; c22 += dz * dz;
        }
        c00 *= 0.1f; c01 *= 0.1f; c02 *= 0.1f; c11 *= 0.1f; c12 *= 0.1f; c22 *= 0.1f;

        // closed-form eigenvalues of symmetric 3x3 (descending)
        float q  = (c00 + c11 + c22) * (1.0f / 3.0f);
        float p1 = c01 * c01 + c02 * c02 + c12 * c12;
        float p2 = (c00 - q) * (c00 - q) + (c11 - q) * (c11 - q) + (c22 - q) * (c22 - q) + 2.0f * p1;
        float e0, e1, e2;
        if (p2 <= 1e-30f) {
            e0 = e1 = e2 = q;
        } else {
            float p = sqrtf(p2 * (1.0f / 6.0f));
            float inv = 1.0f / p;
            float b00 = (c00 - q) * inv, b11 = (c11 - q) * inv, b22 = (c22 - q) * inv;
            float b01 = c01 * inv, b02 = c02 * inv, b12 = c12 * inv;
            float detB = b00 * (b11 * b22 - b12 * b12)
                       - b01 * (b01 * b22 - b12 * b02)
                       + b02 * (b01 * b12 - b11 * b02);
            float r = fminf(1.0f, fmaxf(-1.0f, detB * 0.5f));
            float phi = acosf(r) * (1.0f / 3.0f);
            e0 = q + 2.0f * p * cosf(phi);
            e2 = q + 2.0f * p * cosf(phi + 2.0943951023931953f); // +2pi/3
            e1 = 3.0f * q - e0 - e2;
        }
        float nrm  = sqrtf(e0 * e0 + e1 * e1 + e2 * e2);
        float safe = (e0 > 0.f) ? e0 : 1.0f;
        float r1 = (e0 > 0.f) ? (e0 - e2) / safe : 0.f;
        float r2 = (e0 > 0.f) ? (e1 - e2) / safe : 0.f;
        float r3 = (e0 > 0.f) ? e2 / safe : 0.f;
        float* o = gf + (size_t)task * 7;
        o[0] = e0; o[1] = e1; o[2] = e2; o[3] = nrm; o[4] = r1; o[5] = r2; o[6] = r3;
    }
}

// ---------------------------------------------------------------------------
// Kernel 2: point encoder (3->64->128->256) + feature encoder (7->128->256)
// for the first 64 points of each batch. 512 blocks, 256 threads.
// combined[b,n,:] = [point_feat(256) | fe(256)]
// ---------------------------------------------------------------------------
__global__ __launch_bounds__(256) void encoder_kernel(
    const float* __restrict__ pc, const float* __restrict__ gf,
    const float* __restrict__ pe_w1, const float* __restrict__ pe_b1,
    const float* __restrict__ pe_w2, const float* __restrict__ pe_b2,
    const float* __restrict__ pe_w3, const float* __restrict__ pe_b3,
    const float* __restrict__ fe_w1, const float* __restrict__ fe_b1,
    const float* __restrict__ fe_w2, const float* __restrict__ fe_b2,
    float* __restrict__ combined) {
    int task = blockIdx.x;       // 0..511
    int b = task >> 6, n = task & 63;
    int t = threadIdx.x;         // 0..255
    __shared__ float s1[64];     // h1
    __shared__ float s2[128];    // h2
    __shared__ float sg[128];    // g1 (feature-encoder hidden)

    const float* p = pc + ((size_t)b * NPTS + n) * 3;
    float px = p[0], py = p[1], pz = p[2];
    const float* g = gf + (size_t)task * 7;

    if (t < 64) {
        float a = px * pe_w1[t] + py * pe_w1[64 + t] + pz * pe_w1[128 + t] + pe_b1[t];
        s1[t] = fmaxf(a, 0.f);
    } else if (t >= 128) {       // lanes 128..255 compute g1 (only 7 gf rows nonzero)
        int j = t - 128;
        float a = fe_b1[j];
#pragma unroll
        for (int i = 0; i < 7; ++i) a += g[i] * fe_w1[i * 128 + j];
        sg[j] = fmaxf(a, 0.f);
    }
    __syncthreads();
    if (t < 128) {
        float a = pe_b2[t];
        for (int i = 0; i < 64; ++i) a += s1[i] * pe_w2[i * 128 + t];
        s2[t] = fmaxf(a, 0.f);
    }
    __syncthreads();
    float* cb = combined + (size_t)task * 512;
    {
        float a = pe_b3[t];
        for (int i = 0; i < 128; ++i) a += s2[i] * pe_w3[i * 256 + t];
        cb[t] = a;
    }
    {
        float a = fe_b2[t];
        for (int i = 0; i < 128; ++i) a += sg[i] * fe_w2[i * 256 + t];
        cb[256 + t] = a;
    }
}

// ---------------------------------------------------------------------------
// Kernel 3: map_to_grid — nearest grid cell per point (first-index tie-break),
// last point wins per cell, invalid cells -> 0. Writes grid_feat (ws) and the
// transposed grid_feat_t output. 8 blocks, 256 threads.
// ---------------------------------------------------------------------------
__global__ __launch_bounds__(256) void map_grid_kernel(
    const float* __restrict__ pc, const float* __restrict__ gp,
    const float* __restrict__ combined,
    float* __restrict__ grid_feat, float* __restrict__ out_gft) {
    int b = blockIdx.x;
    int t = threadIdx.x;
    __shared__ int sgidx[64];
    __shared__ int slast[64];
    __shared__ int svalid[64];
    if (t < 64) {
        const float* p = pc + ((size_t)b * NPTS + t) * 3;
        float px = p[0], py = p[1], pz = p[2];
        float best = INFINITY; int bi = 0;
        for (int g = 0; g < GPTS; ++g) {
            float dx = px - gp[g * 3 + 0];
            float dy = py - gp[g * 3 + 1];
            float dz = pz - gp[g * 3 + 2];
            float d = dx * dx + dy * dy + dz * dz;
            if (d < best) { best = d; bi = g; }
        }
        sgidx[t] = bi;
    }
    __syncthreads();
    if (t < 64) {
        int last = -1;
        for (int n = 0; n < 64; ++n)
            if (sgidx[n] == t) last = n;
        slast[t]  = (last < 0) ? 0 : last;
        svalid[t] = (last >= 0) ? 1 : 0;
    }
    __syncthreads();
    for (int i = t; i < 64 * 512; i += 256) {
        int g = i >> 9, c = i & 511;
        float v = svalid[g] ? combined[((size_t)b * 64 + slast[g]) * 512 + c] : 0.f;
        grid_feat[((size_t)b * 64 + g) * 512 + c] = v;
        out_gft[(size_t)b * 512 * 64 + (size_t)c * 64 + g] = v;   // (b, c, g)
    }
}

// ---------------------------------------------------------------------------
// Kernel 4: proc = relu(grid_feat @ rn_w1 + b1) @ rn_w2 + b2 via
// V_WMMA_F32_16X16X4_F32. One block per batch, 16 waves; wave w owns column
// tile w (cols w*16..w*16+15) for all 4 row tiles, sharing one B fragment.
// A frag layout (16x4 f32): lanes 0-15 -> K=k0,k0+1 ; lanes 16-31 -> K=k0+2,k0+3.
// C/D: VGPR r -> M=r (lanes 0-15) / M=8+r (lanes 16-31), N = lane%16.
// ---------------------------------------------------------------------------
__global__ __launch_bounds__(512) void proc_wmma_kernel(
    const float* __restrict__ grid_feat,
    const float* __restrict__ w1, const float* __restrict__ b1,
    const float* __restrict__ w2, const float* __restrict__ b2,
    float* __restrict__ proc_ws, float* __restrict__ out_proct) {
    int b = blockIdx.x;
    int tid = threadIdx.x;
    int wv = tid >> 5;          // wave 0..15 -> column tile
    int lane = tid & 31;
    int lm = lane & 15;
    int khalf = (lane >> 4) << 1;   // 0 or 2
    int col = wv * 16 + lm;

    __shared__ float sh[64 * 256];  // relu hidden, 64KB

    // ---- GEMM 1: (64x512) @ (512x256), K stepped by 4 ----
    const float* A = grid_feat + (size_t)b * 64 * 512;
    v8f acc0 = {}, acc1 = {}, acc2 = {}, acc3 = {};
    for (int k0 = 0; k0 < 512; k0 += 4) {
        int ka = k0 + khalf;
        v2f bf;
        bf.x = w1[(size_t)ka * 256 + col];
        bf.y = w1[(size_t)(ka + 1) * 256 + col];
        v2f a0 = *(const v2f*)(A + (0 * 16 + lm) * 512 + ka);
        v2f a1 = *(const v2f*)(A + (1 * 16 + lm) * 512 + ka);
        v2f a2 = *(const v2f*)(A + (2 * 16 + lm) * 512 + ka);
        v2f a3 = *(const v2f*)(A + (3 * 16 + lm) * 512 + ka);
        acc0 = __builtin_amdgcn_wmma_f32_16x16x4_f32(false, a0, false, bf, (short)0, acc0, false, false);
        acc1 = __builtin_amdgcn_wmma_f32_16x16x4_f32(false, a1, false, bf, (short)0, acc1, false, false);
        acc2 = __builtin_amdgcn_wmma_f32_16x16x4_f32(false, a2, false, bf, (short)0, acc2, false, false);
        acc3 = __builtin_amdgcn_wmma_f32_16x16x4_f32(false, a3, false, bf, (short)0, acc3, false, false);
    }
    float bias1 = b1[col];
#pragma unroll
    for (int r = 0; r < 8; ++r) {
        int m = (lane < 16) ? r : (8 + r);
        sh[(0 * 16 + m) * 256 + col] = fmaxf(acc0[r] + bias1, 0.f);
        sh[(1 * 16 + m) * 256 + col] = fmaxf(acc1[r] + bias1, 0.f);
        sh[(2 * 16 + m) * 256 + col] = fmaxf(acc2[r] + bias1, 0.f);
        sh[(3 * 16 + m) * 256 + col] = fmaxf(acc3[r] + bias1, 0.f);
    }
    __syncthreads();

    // ---- GEMM 2: (64x256) @ (256x256) from LDS ----
    v8f d0 = {}, d1 = {}, d2 = {}, d3 = {};
    for (int k0 = 0; k0 < 256; k0 += 4) {
        int ka = k0 + khalf;
        v2f bf;
        bf.x = w2[(size_t)ka * 256 + col];
        bf.y = w2[(size_t)(ka + 1) * 256 + col];
        v2f a0 = *(const v2f*)(&sh[(0 * 16 + lm) * 256 + ka]);
        v2f a1 = *(const v2f*)(&sh[(1 * 16 + lm) * 256 + ka]);
        v2f a2 = *(const v2f*)(&sh[(2 * 16 + lm) * 256 + ka]);
        v2f a3 = *(const v2f*)(&sh[(3 * 16 + lm) * 256 + ka]);
        d0 = __builtin_amdgcn_wmma_f32_16x16x4_f32(false, a0, false, bf, (short)0, d0, false, false);
        d1 = __builtin_amdgcn_wmma_f32_16x16x4_f32(false, a1, false, bf, (short)0, d1, false, false);
        d2 = __builtin_amdgcn_wmma_f32_16x16x4_f32(false, a2, false, bf, (short)0, d2, false, false);
        d3 = __builtin_amdgcn_wmma_f32_16x16x4_f32(false, a3, false, bf, (short)0, d3, false, false);
    }
    float bias2 = b2[col];
    float* pw = proc_ws  + (size_t)b * 64 * 256;
    float* po = out_proct + (size_t)b * 256 * 64;
#pragma unroll
    for (int r = 0; r < 8; ++r) {
        int m = (lane < 16) ? r : (8 + r);
        float v0 = d0[r] + bias2;
        float v1 = d1[r] + bias2;
        float v2 = d2[r] + bias2;
        float v3 = d3[r] + bias2;
        pw[(0 * 16 + m) * 256 + col] = v0;  po[(size_t)col * 64 + (0 * 16 + m)] = v0;
        pw[(1 * 16 + m) * 256 + col] = v1;  po[(size_t)col * 64 + (1 * 16 + m)] = v1;
        pw[(2 * 16 + m) * 256 + col] = v2;  po[(size_t)col * 64 + (2 * 16 + m)] = v2;
        pw[(3 * 16 + m) * 256 + col] = v3;  po[(size_t)col * 64 + (3 * 16 + m)] = v3;
    }
}

// ---------------------------------------------------------------------------
// Kernel 5: agg = mean over grid cells; three heads + softmax/sigmoid.
// 8 blocks, 128 threads.
// ---------------------------------------------------------------------------
__global__ __launch_bounds__(128) void heads_kernel(
    const float* __restrict__ proc_ws,
    const float* __restrict__ cl_w1, const float* __restrict__ cl_b1,
    const float* __restrict__ cl_w2, const float* __restrict__ cl_b2,
    const float* __restrict__ po_w1, const float* __restrict__ po_b1,
    const float* __restrict__ po_w2, const float* __restrict__ po_b2,
    const float* __restrict__ sz_w1, const float* __restrict__ sz_b1,
    const float* __restrict__ sz_w2, const float* __restrict__ sz_b2,
    float* __restrict__ out) {
    int b = blockIdx.x;
    int t = threadIdx.x;            // 0..127
    __shared__ float sagg[256];
    __shared__ float shid[128];
    __shared__ float sres[128];

    const float* P = proc_ws + (size_t)b * 64 * 256;
    for (int c = t; c < 256; c += 128) {
        float s = 0.f;
        for (int g = 0; g < 64; ++g) s += P[g * 256 + c];
        sagg[c] = s * (1.0f / 64.0f);
    }
    __syncthreads();

    // classifier
    {
        float a = cl_b1[t];
        for (int i = 0; i < 256; ++i) a += sagg[i] * cl_w1[i * 128 + t];
        shid[t] = fmaxf(a, 0.f);
    }
    __syncthreads();
    if (t < 100) {
        float a = cl_b2[t];
        for (int i = 0; i < 128; ++i) a += shid[i] * cl_w2[i * 100 + t];
        sres[t] = a;
    }
    __syncthreads();
    if (t == 0) {
        float mx = -INFINITY;
        for (int j = 0; j < 100; ++j) mx = fmaxf(mx, sres[j]);
        float s = 0.f;
        for (int j = 0; j < 100; ++j) { float e = expf(sres[j] - mx); sres[j] = e; s += e; }
        float inv = 1.0f / s;
        for (int j = 0; j < 100; ++j) out[b * 100 + j] = sres[j] * inv;
    }
    __syncthreads();

    // pose
    {
        float a = po_b1[t];
        for (int i = 0; i < 256; ++i) a += sagg[i] * po_w1[i * 128 + t];
        shid[t] = fmaxf(a, 0.f);
    }
    __syncthreads();
    if (t < 7) {
        float a = po_b2[t];
        for (int i = 0; i < 128; ++i) a += shid[i] * po_w2[i * 7 + t];
        out[800 + b * 7 + t] = a;
    }
    __syncthreads();

    // size
    {
        float a = sz_b1[t];
        for (int i = 0; i < 256; ++i) a += sagg[i] * sz_w1[i * 128 + t];
        shid[t] = fmaxf(a, 0.f);
    }
    __syncthreads();
    if (t < 3) {
        float a = sz_b2[t];
        for (int i = 0; i < 128; ++i) a += shid[i] * sz_w2[i * 3 + t];
        out[856 + b * 3 + t] = 1.0f / (1.0f + expf(-a));
    }
}

// ---------------------------------------------------------------------------
extern "C" void kernel_launch(void* const* d_in, const int* in_sizes, int n_in,
                              void* d_out, int out_size, void* d_ws, size_t ws_size,
                              hipStream_t stream) {
    const float* pc    = (const float*)d_in[0];
    const float* gp    = (const float*)d_in[1];
    const float* pe_w1 = (const float*)d_in[2];
    const float* pe_b1 = (const float*)d_in[3];
    const float* pe_w2 = (const float*)d_in[4];
    const float* pe_b2 = (const float*)d_in[5];
    const float* pe_w3 = (const float*)d_in[6];
    const float* pe_b3 = (const float*)d_in[7];
    const float* fe_w1 = (const float*)d_in[8];
    const float* fe_b1 = (const float*)d_in[9];
    const float* fe_w2 = (const float*)d_in[10];
    const float* fe_b2 = (const float*)d_in[11];
    const float* rn_w1 = (const float*)d_in[12];
    const float* rn_b1 = (const float*)d_in[13];
    const float* rn_w2 = (const float*)d_in[14];
    const float* rn_b2 = (const float*)d_in[15];
    const float* cl_w1 = (const float*)d_in[16];
    const float* cl_b1 = (const float*)d_in[17];
    const float* cl_w2 = (const float*)d_in[18];
    const float* cl_b2 = (const float*)d_in[19];
    const float* po_w1 = (const float*)d_in[20];
    const float* po_b1 = (const float*)d_in[21];
    const float* po_w2 = (const float*)d_in[22];
    const float* po_b2 = (const float*)d_in[23];
    const float* sz_w1 = (const float*)d_in[24];
    const float* sz_b1 = (const float*)d_in[25];
    const float* sz_w2 = (const float*)d_in[26];
    const float* sz_b2 = (const float*)d_in[27];

    float* out = (float*)d_out;
    // output float offsets: class 0(800) | pose 800(56) | size 856(24)
    //                       proc_t 880(131072) | grid_feat_t 131952(262144)
    float* ws = (float*)d_ws;
    float* gf        = ws;            // 8*64*7   = 3584
    float* combined  = ws + 4096;     // 8*64*512 = 262144
    float* grid_feat = ws + 266240;   // 8*64*512 = 262144
    float* proc      = ws + 528384;   // 8*64*256 = 131072

    knn_geom_kernel<<<512, 64, 0, stream>>>(pc, gf);
    encoder_kernel<<<512, 256, 0, stream>>>(pc, gf, pe_w1, pe_b1, pe_w2, pe_b2,
                                            pe_w3, pe_b3, fe_w1, fe_b1, fe_w2, fe_b2,
                                            combined);
    map_grid_kernel<<<8, 256, 0, stream>>>(pc, gp, combined, grid_feat, out + 131952);
    proc_wmma_kernel<<<8, 512, 0, stream>>>(grid_feat, rn_w1, rn_b1, rn_w2, rn_b2,
                                            proc, out + 880);
    heads_kernel<<<8, 128, 0, stream>>>(proc, cl_w1, cl_b1, cl_w2, cl_b2,
                                        po_w1, po_b1, po_w2, po_b2,
                                        sz_w1, sz_b1, sz_w2, sz_b2, out);
}